// MemLinear_53317724012938
// MI455X (gfx1250) — compile-verified
//
#include <hip/hip_runtime.h>
#include <math.h>

typedef float v2f __attribute__((ext_vector_type(2)));
typedef float v4f __attribute__((ext_vector_type(4)));
typedef float v8f __attribute__((ext_vector_type(8)));

#define B_SZ  512
#define IN_SZ 512
#define K_SZ  64
#define V_SZ  64
#define H_SZ  192          // 2K + V
#define M_SZ  2048
#define CIN   768          // IN + H + V == IN + 2K + 2V

// ---------------------------------------------------------------------------
// Kernel 0: materialize inp = [x | relu(hidden) | content]  (512 x 768, f32)
// ---------------------------------------------------------------------------
__global__ void build_inp_kernel(const float* __restrict__ x,
                                 const float* __restrict__ h,
                                 const float* __restrict__ c,
                                 float* __restrict__ inp) {
    int idx = blockIdx.x * blockDim.x + threadIdx.x;
    if (idx >= B_SZ * CIN) return;
    int row = idx / CIN;
    int col = idx - row * CIN;
    float v;
    if (col < IN_SZ) {
        v = x[row * IN_SZ + col];
    } else if (col < IN_SZ + H_SZ) {
        float t = h[row * H_SZ + (col - IN_SZ)];
        v = t > 0.0f ? t : 0.0f;
    } else {
        v = c[row * V_SZ + (col - IN_SZ - H_SZ)];
    }
    inp[idx] = v;
}

// ---------------------------------------------------------------------------
// Kernel 1: o = inp @ W^T + b   via V_WMMA_F32_16X16X4_F32
//   M = 512 (32 tiles), N = 192 (12 tiles), K = 768 (192 steps of 4)
//   One wave32 per 16x16 output tile. 384 waves total.
//
//   A (16x4, f32, 2 VGPR): lanes 0-15 -> M=lane, K={k,k+1};
//                          lanes 16-31 -> M=lane-16, K={k+2,k+3}
//   B (4x16, f32, 2 VGPR): lanes 0-15 -> N=lane, K'={0,1};
//                          lanes 16-31 -> N=lane-16, K'={2,3}
//   B[k', n] = W[n0+n, k+k']  (since o = inp @ W^T)
//   C/D (16x16 f32, 8 VGPR): lanes 0-15 rows 0-7, lanes 16-31 rows 8-15
// ---------------------------------------------------------------------------
__global__ __launch_bounds__(128) void gemm_wmma_kernel(
        const float* __restrict__ inp, const float* __restrict__ W,
        const float* __restrict__ bias, float* __restrict__ o) {
    const int lane = threadIdx.x & 31;
    const int wave = blockIdx.x * (blockDim.x >> 5) + (threadIdx.x >> 5);
    const int nt = wave % (H_SZ / 16);   // 0..11
    const int mt = wave / (H_SZ / 16);   // 0..31
    const int m0 = mt * 16;
    const int n0 = nt * 16;

    const int half = lane >> 4;          // 0 or 1
    const int l15  = lane & 15;

    const float* aRow = inp + (size_t)(m0 + l15) * CIN + 2 * half;
    const float* bRow = W   + (size_t)(n0 + l15) * CIN + 2 * half;

    v8f acc = {};
    #pragma unroll 4
    for (int k = 0; k < CIN; k += 4) {
        v2f a = *(const v2f*)(aRow + k);
        v2f b = *(const v2f*)(bRow + k);
        acc = __builtin_amdgcn_wmma_f32_16x16x4_f32(
            /*neg_a=*/false, a, /*neg_b=*/false, b,
            /*c_mod=*/(short)0, acc, /*reuse_a=*/false, /*reuse_b=*/false);
    }

    const int col = n0 + l15;
    const float bv = bias[col];
    #pragma unroll
    for (int r = 0; r < 8; ++r) {
        const int row = m0 + r + 8 * half;
        o[(size_t)row * H_SZ + col] = acc[r] + bv;
    }
}

// ---------------------------------------------------------------------------
// Kernel 2: per-batch  scores -> softmax -> content_new   (pure streaming)
//   One 256-thread block per batch; streams 1 MB (key_mem + value_mem slices)
//   as non-temporal b128 loads (512 B/wave/instr). Softmax staged in LDS,
//   attn kept as 16-byte vectors so LDS traffic is ds_*_b128.
//
//   m-ownership (scores): thread t owns m = 4t..4t+3  and  1024+4t..1024+4t+3
// ---------------------------------------------------------------------------
__global__ __launch_bounds__(256) void attn_kernel(
        const float* __restrict__ o,
        const float* __restrict__ key_mem,
        const float* __restrict__ value_mem,
        float* __restrict__ content_out) {
    const int b = blockIdx.x;
    const int t = threadIdx.x;          // 0..255

    __shared__ float q_sh[K_SZ];
    __shared__ v4f   attn4_sh[M_SZ / 4];   // 2048 floats, 16B aligned
    __shared__ float red_sh[256];

    if (t < K_SZ) q_sh[t] = o[(size_t)b * H_SZ + t];
    __syncthreads();

    // ---- scores[m] = sum_k key_mem[b,k,m] * q[k] ; two float4 groups/thread
    const float* km = key_mem + (size_t)b * K_SZ * M_SZ;
    v4f a0 = {0.0f, 0.0f, 0.0f, 0.0f};
    v4f a1 = {0.0f, 0.0f, 0.0f, 0.0f};

    for (int k = 0; k < K_SZ; ++k) {
        const float qk = q_sh[k];
        const v4f* row4 = (const v4f*)(km + (size_t)k * M_SZ);
        if (k + 1 < K_SZ)
            __builtin_prefetch((const float*)(row4 + M_SZ / 4) + 4 * t, 0, 1);
        v4f r0 = __builtin_nontemporal_load(row4 + t);
        v4f r1 = __builtin_nontemporal_load(row4 + t + 256);
        a0 += r0 * qk;      // contracts to v_fma under HIP default fp-contract
        a1 += r1 * qk;
    }

    // ---- softmax over 2048 scores (max, exp, sum, normalize) ----
    float mx = a0[0];
    #pragma unroll
    for (int j = 1; j < 4; ++j) mx = fmaxf(mx, a0[j]);
    #pragma unroll
    for (int j = 0; j < 4; ++j) mx = fmaxf(mx, a1[j]);
    red_sh[t] = mx;
    __syncthreads();
    for (int s = 128; s > 0; s >>= 1) {
        if (t < s) red_sh[t] = fmaxf(red_sh[t], red_sh[t + s]);
        __syncthreads();
    }
    mx = red_sh[0];
    __syncthreads();

    v4f e0, e1;
    float sum = 0.0f;
    #pragma unroll
    for (int j = 0; j < 4; ++j) { e0[j] = __expf(a0[j] - mx); sum += e0[j]; }
    #pragma unroll
    for (int j = 0; j < 4; ++j) { e1[j] = __expf(a1[j] - mx); sum += e1[j]; }
    red_sh[t] = sum;
    __syncthreads();
    for (int s = 128; s > 0; s >>= 1) {
        if (t < s) red_sh[t] += red_sh[t + s];
        __syncthreads();
    }
    const float inv = 1.0f / red_sh[0];
    __syncthreads();

    attn4_sh[t]       = e0 * inv;     // ds_store_b128
    attn4_sh[t + 256] = e1 * inv;
    __syncthreads();

    // ---- content_new[v] = sum_m value_mem[b,v,m] * attn[m] ----
    // 8 waves x 8 rows each; lane l reads float4 at m-chunk l+32*i
    const float* vm = value_mem + (size_t)b * V_SZ * M_SZ;
    const int wave = t >> 5;
    const int lane = t & 31;
    for (int v = wave * 8; v < wave * 8 + 8; ++v) {
        const v4f* row4 = (const v4f*)(vm + (size_t)v * M_SZ);
        v4f sv = {0.0f, 0.0f, 0.0f, 0.0f};
        for (int i = lane; i < M_SZ / 4; i += 32) {
            v4f r = __builtin_nontemporal_load(row4 + i);   // b128, 512B/wave
            sv += r * attn4_sh[i];                          // ds_load_b128
        }
        float s = sv[0] + sv[1] + sv[2] + sv[3];
        #pragma unroll
        for (int off = 16; off > 0; off >>= 1)
            s += __shfl_down(s, off, 32);
        if (lane == 0) content_out[(size_t)b * V_SZ + v] = s;
    }
}

// ---------------------------------------------------------------------------
extern "C" void kernel_launch(void* const* d_in, const int* in_sizes, int n_in,
                              void* d_out, int out_size, void* d_ws, size_t ws_size,
                              hipStream_t stream) {
    const float* x       = (const float*)d_in[0];   // (512, 512)
    const float* hidden  = (const float*)d_in[1];   // (512, 192)
    const float* content = (const float*)d_in[2];   // (512, 64)
    const float* key_mem = (const float*)d_in[3];   // (512, 64, 2048)
    const float* val_mem = (const float*)d_in[4];   // (512, 64, 2048)
    const float* W       = (const float*)d_in[5];   // (192, 768)
    const float* bias    = (const float*)d_in[6];   // (192,)

    float* out_o = (float*)d_out;                   // (512, 192)
    float* out_c = out_o + (size_t)B_SZ * H_SZ;     // (512, 64)

    float* ws_inp = (float*)d_ws;                   // (512, 768) staging

    // Phase 0: build concatenated input
    {
        const int n = B_SZ * CIN;
        build_inp_kernel<<<(n + 255) / 256, 256, 0, stream>>>(x, hidden, content, ws_inp);
    }
    // Phase 1: o = inp @ W^T + b  (WMMA f32 16x16x4; 384 waves, 4 waves/block)
    gemm_wmma_kernel<<<96, 128, 0, stream>>>(ws_inp, W, bias, out_o);
    // Phase 2: streaming attention (one block per batch)
    attn_kernel<<<B_SZ, 256, 0, stream>>>(out_o, key_mem, val_mem, out_c);
}